// CoordDeformableC3D_53300544143734
// MI455X (gfx1250) — compile-verified
//
#include <hip/hip_runtime.h>

// CoordDeformableC3D for MI455X (gfx1250, wave32).
//
// Pre-pass packs (one-time, ~20 MB scratch, all L2-resident):
//   xT[m][68]      : x channel-transposed + coord channels 64..66 + zero pad 67
//   wP[27][81][68] : offset-conv weights, tap-major, channel-contiguous (pad 0)
//   wD[27][64][64] : deform weights,     tap-major, channel-contiguous
//
// Stage 1: offset conv as implicit GEMM on V_WMMA_F32_16X16X4_F32 (exact fp32),
//   K-order = tap-major so each WMMA chunk is 4 adjacent channels of one tap:
//   B = one b64 gather per N-tile, A = ds_load_b64 from an LDS-staged slab
//   shared by all 4 waves (double-buffered, 44 KB of the 320 KB WGP LDS).
// Stage 2: fused trilinear-gather + GEMM, same LDS slab trick for weights,
//   gathers pull 2 channels per b64 from xT. Corner indices/weights computed
//   once per (tap, position) and reused across all 64 channels.

typedef __attribute__((ext_vector_type(2))) float v2f;
typedef __attribute__((ext_vector_type(8))) float v8f;

#define DD 8
#define HH 64
#define WW 64
#define MTOT (DD * HH * WW)   // 32768 positions
#define CIN 64
#define CP 68                 // 64 data + 3 coord + 1 zero-pad channels
#define K3 27
#define KTOT (67 * K3)        // 1809 (original w_off K extent)
#define OCOFF 81
#define COUT 64

#define WP_SLAB (OCOFF * CP)  // 5508 floats per tap (stage 1)
#define WD_SLAB (COUT * CIN)  // 4096 floats per tap (stage 2)

__device__ __forceinline__ v8f wmma_f32(v2f a, v2f b, v8f c) {
  return __builtin_amdgcn_wmma_f32_16x16x4_f32(false, a, false, b, (short)0, c,
                                               false, false);
}

// ---------------- pre-pass: xT[m][68] = { x[c][m], coords(m), 0 }
__global__ __launch_bounds__(256) void build_xt_kernel(
    const float* __restrict__ x, float* __restrict__ xT) {
  const int t = blockIdx.x * 256 + threadIdx.x;
  if (t >= MTOT * CP) return;
  const int m = t / CP;
  const int c = t - m * CP;
  float v;
  if (c < CIN) {
    v = x[c * MTOT + m];
  } else if (c == 64) {
    v = -1.0f + (float)(m >> 12) * (2.0f / 7.0f);          // z coord
  } else if (c == 65) {
    v = -1.0f + (float)((m >> 6) & 63) * (2.0f / 63.0f);   // y coord
  } else if (c == 66) {
    v = -1.0f + (float)(m & 63) * (2.0f / 63.0f);          // x coord
  } else {
    v = 0.0f;                                              // pad channel
  }
  xT[t] = v;
}

// ---------------- pre-pass: wP[kidx][oc][c] from w_off[oc][c][kidx]
__global__ __launch_bounds__(256) void pack_woff_kernel(
    const float* __restrict__ w_off, float* __restrict__ wP) {
  const int t = blockIdx.x * 256 + threadIdx.x;
  if (t >= K3 * WP_SLAB) return;
  const int kidx = t / WP_SLAB;
  const int r = t - kidx * WP_SLAB;
  const int oc = r / CP;
  const int c = r - oc * CP;
  wP[t] = (c < 67) ? w_off[oc * KTOT + c * K3 + kidx] : 0.0f;
}

// ---------------- pre-pass: wD[k][oc][c] from w_def[oc][c][k]
__global__ __launch_bounds__(256) void pack_wdef_kernel(
    const float* __restrict__ w_def, float* __restrict__ wD) {
  const int t = blockIdx.x * 256 + threadIdx.x;
  if (t >= K3 * WD_SLAB) return;
  const int c = t & 63;
  const int oc = (t >> 6) & 63;
  const int k = t >> 12;
  wD[t] = w_def[(oc * CIN + c) * K3 + k];
}

// cooperative contiguous slab copy: global -> LDS, b128 granularity
__device__ __forceinline__ void copy_slab(float* __restrict__ dst,
                                          const float* __restrict__ src,
                                          int nfloat4, int tid) {
  const float4* s = (const float4*)src;
  float4* d = (float4*)dst;
  for (int i = tid; i < nfloat4; i += 128) d[i] = s[i];
}

// ---------------- Kernel 1: offsets = conv3d(concat(x, coords), w_off) + b_off
// grid: 256 blocks * 4 waves; wave = one (z,y) row half = 2 N-tiles of 16.
__global__ __launch_bounds__(128) void offsets_conv_kernel(
    const float* __restrict__ xT, const float* __restrict__ wP,
    const float* __restrict__ b_off, float* __restrict__ offs) {
  __shared__ float wlds[2][WP_SLAB];
  const int tid = threadIdx.x;
  const int lane = tid & 31;
  const int wave = tid >> 5;
  const int n = lane & 15;      // WMMA column
  const int half = lane >> 4;   // selects K-slots {0,1} vs {2,3} of a chunk
  const int zy = blockIdx.x * 2 + (wave >> 1);  // 0..511 (pair shares z)
  const int z = zy >> 6;
  const int y = zy & 63;
  const int xpos0 = (wave & 1) * 32 + n;        // N-tile A; N-tile B = +16

  v8f accA[6], accB[6];
#pragma unroll
  for (int i = 0; i < 6; ++i)
#pragma unroll
    for (int e = 0; e < 8; ++e) {
      accA[i][e] = 0.0f;
      accB[i][e] = 0.0f;
    }

  copy_slab(wlds[0], wP, WP_SLAB / 4, tid);
  __syncthreads();

  for (int kidx = 0; kidx < K3; ++kidx) {
    // prefetch next tap's weight slab into the other buffer (used slab from
    // kidx-1 is safe to overwrite: all waves passed the previous barrier)
    if (kidx + 1 < K3)
      copy_slab(wlds[(kidx + 1) & 1], wP + (kidx + 1) * WP_SLAB, WP_SLAB / 4,
                tid);

    // tap geometry (once per tap, shared by all 17 channel-chunks)
    const int dz = kidx / 9;
    const int rem = kidx - dz * 9;
    const int dy = rem / 3;
    const int dx = rem - dy * 3;
    const int zz = z + dz - 1, yy = y + dy - 1;
    const int xxA = xpos0 + dx - 1, xxB = xxA + 16;
    const bool okzy = ((unsigned)zz < DD) & ((unsigned)yy < HH);
    const int base =
        (min(max(zz, 0), DD - 1) * HH + min(max(yy, 0), HH - 1)) * WW;
    const float mA = (okzy & ((unsigned)xxA < WW)) ? 1.0f : 0.0f;
    const float mB = (okzy & ((unsigned)xxB < WW)) ? 1.0f : 0.0f;
    const float* __restrict__ xrA =
        xT + (size_t)(base + min(max(xxA, 0), WW - 1)) * CP;
    const float* __restrict__ xrB =
        xT + (size_t)(base + min(max(xxB, 0), WW - 1)) * CP;
    const float* __restrict__ wl = wlds[kidx & 1];

#pragma unroll
    for (int cc = 0; cc < 17; ++cc) {  // 68 channels / 4 per chunk
      const int c0 = cc * 4 + half * 2;
      const v2f gA = *(const v2f*)(xrA + c0);  // one b64 gather per N-tile
      const v2f gB = *(const v2f*)(xrB + c0);
      v2f bA, bB;
      bA.x = gA.x * mA;
      bA.y = gA.y * mA;
      bB.x = gB.x * mB;
      bB.y = gB.y * mB;
#pragma unroll
      for (int mt = 0; mt < 6; ++mt) {
        // rows oc>=81 clamp to 80: garbage D rows, never stored
        const int oc = min(mt * 16 + n, OCOFF - 1);
        const v2f a = *(const v2f*)(wl + oc * CP + c0);  // ds_load_b64
        accA[mt] = wmma_f32(a, bA, accA[mt]);
        accB[mt] = wmma_f32(a, bB, accB[mt]);
      }
    }
    __syncthreads();
  }

  const int mbase = (zy << 6) + xpos0;
#pragma unroll
  for (int mt = 0; mt < 6; ++mt)
#pragma unroll
    for (int r = 0; r < 8; ++r) {
      const int oc = mt * 16 + r + 8 * half;  // D-fragment row mapping
      if (oc < OCOFF) {
        const float bias = b_off[oc];
        offs[oc * MTOT + mbase] = accA[mt][r] + bias;
        offs[oc * MTOT + mbase + 16] = accB[mt][r] + bias;
      }
    }
}

// ---------------- Kernel 2: fused trilinear gather + deform GEMM
// out[oc, m] = sum_{c,k} w_def[oc,c,k] * interp(x[c], base(m)+tap(k)+off(k,m))
__global__ __launch_bounds__(128) void deform_gemm_kernel(
    const float* __restrict__ xT, const float* __restrict__ offs,
    const float* __restrict__ wD, const float* __restrict__ b_def,
    float* __restrict__ out) {
  __shared__ float wlds[2][WD_SLAB];
  const int tid = threadIdx.x;
  const int lane = tid & 31;
  const int wave = tid >> 5;
  const int n = lane & 15;
  const int half = lane >> 4;
  const int m = (blockIdx.x * 4 + wave) * 16 + n;
  const int z = m >> 12;
  const int y = (m >> 6) & 63;
  const int xp = m & 63;

  v8f acc[4];
#pragma unroll
  for (int i = 0; i < 4; ++i)
#pragma unroll
    for (int e = 0; e < 8; ++e) acc[i][e] = 0.0f;

  copy_slab(wlds[0], wD, WD_SLAB / 4, tid);
  __syncthreads();

  for (int k = 0; k < K3; ++k) {
    if (k + 1 < K3)
      copy_slab(wlds[(k + 1) & 1], wD + (k + 1) * WD_SLAB, WD_SLAB / 4, tid);

    const int dz = k / 9 - 1;
    const int dy = (k / 3) % 3 - 1;
    const int dx = k % 3 - 1;
    // sampling point p = base + tap + learned offset
    const float pz = (float)(z + dz) + offs[(k * 3 + 0) * MTOT + m];
    const float py = (float)(y + dy) + offs[(k * 3 + 1) * MTOT + m];
    const float px = (float)(xp + dx) + offs[(k * 3 + 2) * MTOT + m];
    const float z0f = floorf(pz), y0f = floorf(py), x0f = floorf(px);
    const float lz = pz - z0f, ly = py - y0f, lx = px - x0f;
    const int z0 = (int)z0f, y0 = (int)y0f, x0 = (int)x0f;

    // 8 corner row pointers + weights, computed ONCE per (tap, position),
    // reused across all 64 channels
    int widx[8];
    float wv[8];
#pragma unroll
    for (int j = 0; j < 8; ++j) {
      const int jz = (j >> 2) & 1, jy = (j >> 1) & 1, jx = j & 1;
      const int zi = z0 + jz, yi = y0 + jy, xi = x0 + jx;
      const bool ok =
          ((unsigned)zi < DD) & ((unsigned)yi < HH) & ((unsigned)xi < WW);
      const int zc = min(max(zi, 0), DD - 1);
      const int yc = min(max(yi, 0), HH - 1);
      const int xc = min(max(xi, 0), WW - 1);
      widx[j] = ((zc * HH + yc) * WW + xc) * CP;  // row offset into xT
      const float w = (jz ? lz : 1.0f - lz) * (jy ? ly : 1.0f - ly) *
                      (jx ? lx : 1.0f - lx);
      wv[j] = ok ? w : 0.0f;
    }

    const float* __restrict__ wl = wlds[k & 1];
#pragma unroll 4
    for (int cb = 0; cb < CIN; cb += 4) {
      const int c0 = cb + half * 2;  // this lane's two channel slots (even)
      v2f b;
      b.x = 0.0f;
      b.y = 0.0f;
#pragma unroll
      for (int j = 0; j < 8; ++j) {
        const v2f g = *(const v2f*)(xT + widx[j] + c0);  // 2 channels / b64
        b.x += wv[j] * g.x;
        b.y += wv[j] * g.y;
      }
#pragma unroll
      for (int mt = 0; mt < 4; ++mt) {
        const int oc = mt * 16 + n;
        const v2f a = *(const v2f*)(wl + oc * CIN + c0);  // ds_load_b64
        acc[mt] = wmma_f32(a, b, acc[mt]);
      }
    }
    __syncthreads();
  }

#pragma unroll
  for (int mt = 0; mt < 4; ++mt)
#pragma unroll
    for (int r = 0; r < 8; ++r) {
      const int oc = mt * 16 + r + 8 * half;
      out[oc * MTOT + m] = acc[mt][r] + b_def[oc];
    }
}

extern "C" void kernel_launch(void* const* d_in, const int* in_sizes, int n_in,
                              void* d_out, int out_size, void* d_ws,
                              size_t ws_size, hipStream_t stream) {
  const float* x = (const float*)d_in[0];       // [1,64,8,64,64]
  const float* w_off = (const float*)d_in[1];   // [81,67,3,3,3]
  const float* b_off = (const float*)d_in[2];   // [81]
  const float* w_def = (const float*)d_in[3];   // [64,64,3,3,3]
  const float* b_def = (const float*)d_in[4];   // [64]
  float* out = (float*)d_out;                   // [1,64,8,64,64]

  // scratch layout (floats): offs | xT | wP | wD   (~20.6 MB total)
  float* offs = (float*)d_ws;                      // 81*32768
  float* xT = offs + (size_t)OCOFF * MTOT;         // 32768*68
  float* wP = xT + (size_t)MTOT * CP;              // 27*81*68
  float* wD = wP + (size_t)K3 * WP_SLAB;           // 27*64*64

  build_xt_kernel<<<(MTOT * CP + 255) / 256, 256, 0, stream>>>(x, xT);
  pack_woff_kernel<<<(K3 * WP_SLAB + 255) / 256, 256, 0, stream>>>(w_off, wP);
  pack_wdef_kernel<<<(K3 * WD_SLAB + 255) / 256, 256, 0, stream>>>(w_def, wD);
  offsets_conv_kernel<<<256, 128, 0, stream>>>(xT, wP, b_off, offs);
  deform_gemm_kernel<<<512, 128, 0, stream>>>(xT, offs, wD, b_def, out);
}